// SparseMM_21569325761081
// MI455X (gfx1250) — compile-verified
//
#include <hip/hip_runtime.h>
#include <hip/hip_bf16.h>
#include <stdint.h>

#define D_FEAT 128
#define WAVE 32
#define BATCH 32             // edges per wave-batch
#define META_BYTES 640       // 256B src(i64) + 256B dst(i64) + 128B att(f32)
#define WAVES_PER_BLOCK 8
#define BLOCK (WAVES_PER_BLOCK * WAVE)

// LDS byte offset of a generic pointer into __shared__ (low 32 bits of the
// LDS-aperture flat address are the LDS address per CDNA5 aperture mapping).
__device__ __forceinline__ unsigned lds_off(const void* p) {
    return (unsigned)(unsigned long long)(uintptr_t)p;
}

// Guaranteed-native no-return fp32 atomic add (resolves at L2).
__device__ __forceinline__ void atomic_add_f32(float* p, float v) {
    asm volatile("global_atomic_add_f32 %0, %1, off" :: "v"(p), "v"(v) : "memory");
}

// ============================================================================
// CSR pull-based pipeline (primary path)
// ============================================================================

// Stage one 32-edge CSR chunk's (dst, att) into LDS via async global->LDS.
__device__ __forceinline__ void stage_chunk(const int* csr_dst, const float* csr_att,
                                            unsigned chunk, unsigned end,
                                            unsigned char* buf, int lane) {
    unsigned idx = chunk + (unsigned)lane;
    if (idx >= end) idx = end - 1;           // clamp (end>=1 since chunk<end)
    const char* gd = (const char*)(csr_dst + idx);
    unsigned ld = lds_off(buf) + (unsigned)lane * 4u;
    asm volatile("global_load_async_to_lds_b32 %0, %1, off th:TH_LOAD_NT"
                 :: "v"(ld), "v"(gd) : "memory");
    const char* ga = (const char*)(csr_att + idx);
    unsigned la = lds_off(buf) + 128u + (unsigned)lane * 4u;
    asm volatile("global_load_async_to_lds_b32 %0, %1, off th:TH_LOAD_NT"
                 :: "v"(la), "v"(ga) : "memory");
}

// ---- gather kernel: one wave per node, FMA accumulation, zero fp atomics ----
__global__ void __launch_bounds__(BLOCK)
spmm_gather_kernel(const unsigned* __restrict__ off,
                   const int* __restrict__ csr_dst,
                   const float* __restrict__ csr_att,
                   const float* __restrict__ X,
                   float* __restrict__ out, int N) {
    __shared__ unsigned char smem[WAVES_PER_BLOCK * 2 * 256];
    const int lane = threadIdx.x & (WAVE - 1);
    const int wib  = threadIdx.x >> 5;
    const int nWaves = gridDim.x * WAVES_PER_BLOCK;
    const int gwave  = blockIdx.x * WAVES_PER_BLOCK + wib;
    unsigned char* buf0 = smem + (wib * 2 + 0) * 256;
    unsigned char* buf1 = smem + (wib * 2 + 1) * 256;

    for (int n = gwave; n < N; n += nWaves) {
        unsigned beg = (n == 0) ? 0u : off[n - 1];   // post-fill: off[] = inclusive ends
        unsigned end = off[n];
        float4 acc; acc.x = 0.f; acc.y = 0.f; acc.z = 0.f; acc.w = 0.f;

        if (beg < end) {
            // software-pipelined chunk staging: next chunk's async loads are in
            // flight while this chunk's 32-edge gather/FMA loop runs
            stage_chunk(csr_dst, csr_att, beg, end, buf0, lane);
            int parity = 0;
            for (unsigned chunk = beg; chunk < end; chunk += WAVE, parity ^= 1) {
                unsigned nxt = chunk + WAVE;
                unsigned char* cur = parity ? buf1 : buf0;
                unsigned char* stg = parity ? buf0 : buf1;
                if (nxt < end) {
                    stage_chunk(csr_dst, csr_att, nxt, end, stg, lane);
                    asm volatile("s_wait_asynccnt 0x2" ::: "memory");
                } else {
                    asm volatile("s_wait_asynccnt 0x0" ::: "memory");
                }

                const int*   bD = (const int*)cur;
                const float* bA = (const float*)(cur + 128);

                // each lane prefetches one full 512B X-row of this chunk
                {
                    int ddl = bD[lane];
                    const char* p = (const char*)(X + (size_t)ddl * D_FEAT);
                    __builtin_prefetch(p +   0, 0, 3);
                    __builtin_prefetch(p + 128, 0, 3);
                    __builtin_prefetch(p + 256, 0, 3);
                    __builtin_prefetch(p + 384, 0, 3);
                }

                int cnt = (int)((end - chunk) < (unsigned)WAVE ? (end - chunk)
                                                               : (unsigned)WAVE);
#pragma unroll 4
                for (int i = 0; i < cnt; ++i) {
                    int   dd = bD[i];                // LDS broadcast reads
                    float a  = bA[i];
                    const float4 v = *(const float4*)(X + (size_t)dd * D_FEAT + lane * 4);
                    acc.x = fmaf(a, v.x, acc.x);
                    acc.y = fmaf(a, v.y, acc.y);
                    acc.z = fmaf(a, v.z, acc.z);
                    acc.w = fmaf(a, v.w, acc.w);
                }
            }
        }
        *(float4*)(out + (size_t)n * D_FEAT + lane * 4) = acc;  // coalesced 512B/row
    }
}

// ---- zero u32 workspace ----
__global__ void spmm_zero_u32(unsigned* __restrict__ p, int n) {
    int i = blockIdx.x * blockDim.x + threadIdx.x;
    int stride = gridDim.x * blockDim.x;
    for (; i < n; i += stride) p[i] = 0u;
}

// ---- histogram of src ----
__global__ void spmm_hist_kernel(const long long* __restrict__ edges,
                                 unsigned* __restrict__ cnt, long long E) {
    long long i = (long long)blockIdx.x * blockDim.x + threadIdx.x;
    long long stride = (long long)gridDim.x * blockDim.x;
    for (; i < E; i += stride) {
        int s = (int)edges[i];
        atomicAdd(&cnt[s], 1u);   // native global_atomic_add_u32
    }
}

// ---- single-block in-place exclusive scan (shfl-based, 3 barriers/chunk) ----
#define SCAN_T 1024
__global__ void __launch_bounds__(SCAN_T)
spmm_scan_kernel(unsigned* __restrict__ data, int N) {
    __shared__ unsigned sWave[SCAN_T / WAVE];   // 32 wave totals
    const int tid  = threadIdx.x;
    const int lane = tid & (WAVE - 1);
    const int wib  = tid >> 5;
    unsigned carry = 0u;                         // uniform across all threads
    const int CH = SCAN_T * 4;

    for (int base = 0; base < N; base += CH) {
        int i0 = base + tid * 4;
        unsigned v0 = (i0 + 0 < N) ? data[i0 + 0] : 0u;
        unsigned v1 = (i0 + 1 < N) ? data[i0 + 1] : 0u;
        unsigned v2 = (i0 + 2 < N) ? data[i0 + 2] : 0u;
        unsigned v3 = (i0 + 3 < N) ? data[i0 + 3] : 0u;
        unsigned t0 = v0, t1 = t0 + v1, t2 = t1 + v2, t3 = t2 + v3;

        unsigned inc = t3;                       // intra-wave inclusive scan
        for (int d = 1; d < WAVE; d <<= 1) {
            unsigned t = __shfl_up(inc, d);
            if (lane >= d) inc += t;
        }
        __syncthreads();                         // protect sWave from prev readers
        if (lane == WAVE - 1) sWave[wib] = inc;
        __syncthreads();
        if (wib == 0) {                          // wave 0 scans the 32 wave totals
            unsigned w = sWave[lane];
            for (int d = 1; d < WAVE; d <<= 1) {
                unsigned t = __shfl_up(w, d);
                if (lane >= d) w += t;
            }
            sWave[lane] = w;                     // inclusive wave totals
        }
        __syncthreads();
        unsigned waveBase = (wib == 0) ? 0u : sWave[wib - 1];
        unsigned b0 = carry + waveBase + (inc - t3);  // exclusive before this thread
        if (i0 + 0 < N) data[i0 + 0] = b0;
        if (i0 + 1 < N) data[i0 + 1] = b0 + t0;
        if (i0 + 2 < N) data[i0 + 2] = b0 + t1;
        if (i0 + 3 < N) data[i0 + 3] = b0 + t2;
        carry += sWave[SCAN_T / WAVE - 1];       // uniform update, no extra barrier
    }
}

// ---- CSR fill: off[] incremented in place -> becomes inclusive ends ----
__global__ void spmm_fill_kernel(const long long* __restrict__ edges,
                                 const float* __restrict__ att,
                                 unsigned* __restrict__ off,
                                 int* __restrict__ csr_dst,
                                 float* __restrict__ csr_att, long long E) {
    long long i = (long long)blockIdx.x * blockDim.x + threadIdx.x;
    long long stride = (long long)gridDim.x * blockDim.x;
    for (; i < E; i += stride) {
        int s = (int)edges[i];
        unsigned slot = atomicAdd(&off[s], 1u);
        csr_dst[slot] = (int)edges[E + i];
        csr_att[slot] = att[i];
    }
}

// ============================================================================
// Fallback: edge-parallel atomic scatter (used only if ws_size is too small)
// ============================================================================

__device__ __forceinline__ void stage_batch(const long long* edges,
                                            const float* att,
                                            long long E, long long base,
                                            unsigned char* buf, int lane) {
    const char* gmeta;
    if (lane < 16)
        gmeta = (const char*)edges + base * 8 + (long long)lane * 16;      // src
    else
        gmeta = (const char*)edges + E * 8 + base * 8 + (long long)(lane - 16) * 16; // dst
    unsigned l0 = lds_off(buf) + (unsigned)lane * 16u;
    asm volatile("global_load_async_to_lds_b128 %0, %1, off th:TH_LOAD_NT"
                 :: "v"(l0), "v"(gmeta) : "memory");
    const char* gatt = (const char*)att + (base + lane) * 4;
    unsigned l1 = lds_off(buf) + 512u + (unsigned)lane * 4u;
    asm volatile("global_load_async_to_lds_b32 %0, %1, off th:TH_LOAD_NT"
                 :: "v"(l1), "v"(gatt) : "memory");
}

__device__ __forceinline__ void process_batch(const unsigned char* buf,
                                              const float* __restrict__ X,
                                              float* __restrict__ out, int lane) {
    const unsigned long long* sS = (const unsigned long long*)buf;
    const unsigned long long* sD = (const unsigned long long*)(buf + 256);
    const float*              sA = (const float*)(buf + 512);
    {
        long long dd = (long long)sD[lane];
        const char* p = (const char*)(X + dd * (long long)D_FEAT);
        __builtin_prefetch(p, 0, 3);       __builtin_prefetch(p + 128, 0, 3);
        __builtin_prefetch(p + 256, 0, 3); __builtin_prefetch(p + 384, 0, 3);
    }
#pragma unroll 4
    for (int e = 0; e < BATCH; ++e) {
        int   s  = (int)sS[e];
        int   dd = (int)sD[e];
        float a  = sA[e];
        const float4 v = *(const float4*)(X + (size_t)dd * D_FEAT + lane * 4);
        float* o = out + (size_t)s * D_FEAT + lane * 4;
        asm volatile("global_atomic_add_f32 %0, %1, off"           :: "v"(o), "v"(a * v.x) : "memory");
        asm volatile("global_atomic_add_f32 %0, %1, off offset:4"  :: "v"(o), "v"(a * v.y) : "memory");
        asm volatile("global_atomic_add_f32 %0, %1, off offset:8"  :: "v"(o), "v"(a * v.z) : "memory");
        asm volatile("global_atomic_add_f32 %0, %1, off offset:12" :: "v"(o), "v"(a * v.w) : "memory");
    }
}

__global__ void __launch_bounds__(BLOCK)
spmm_scatter_kernel(const long long* __restrict__ edges,
                    const float* __restrict__ att,
                    const float* __restrict__ X,
                    float* __restrict__ out, long long E) {
    __shared__ unsigned char smem[WAVES_PER_BLOCK * 2 * META_BYTES];
    const int lane = threadIdx.x & (WAVE - 1);
    const int wib  = threadIdx.x >> 5;
    const long long nWaves = (long long)gridDim.x * WAVES_PER_BLOCK;
    const long long gwave  = (long long)blockIdx.x * WAVES_PER_BLOCK + wib;
    const long long nFull  = E / BATCH;
    unsigned char* buf0 = smem + (wib * 2 + 0) * META_BYTES;
    unsigned char* buf1 = smem + (wib * 2 + 1) * META_BYTES;

    long long b = gwave;
    if (b < nFull) {
        stage_batch(edges, att, E, b * BATCH, buf0, lane);
        int parity = 0;
        for (; b < nFull; b += nWaves) {
            long long nb = b + nWaves;
            unsigned char* cur = parity ? buf1 : buf0;
            unsigned char* nxt = parity ? buf0 : buf1;
            if (nb < nFull) {
                stage_batch(edges, att, E, nb * BATCH, nxt, lane);
                asm volatile("s_wait_asynccnt 0x2" ::: "memory");
            } else {
                asm volatile("s_wait_asynccnt 0x0" ::: "memory");
            }
            process_batch(cur, X, out, lane);
            parity ^= 1;
        }
    }
    if (gwave == 0) {                       // generic tail (E % 32 != 0)
        long long tail = nFull * BATCH;
        long long e = tail + lane;
        if (e < E) {
            long long s  = edges[e];
            long long dd = edges[E + e];
            float a = att[e];
            const float* xr = X + dd * (long long)D_FEAT;
            float* o = out + s * (long long)D_FEAT;
            for (int d = 0; d < D_FEAT; ++d) atomic_add_f32(o + d, a * xr[d]);
        }
    }
}

__global__ void spmm_zero_kernel(float4* __restrict__ out4, int n4) {
    int i = blockIdx.x * blockDim.x + threadIdx.x;
    int stride = gridDim.x * blockDim.x;
    float4 z; z.x = 0.f; z.y = 0.f; z.z = 0.f; z.w = 0.f;
    for (; i < n4; i += stride) out4[i] = z;
}

// ---------------------------------------------------------------- launcher
extern "C" void kernel_launch(void* const* d_in, const int* in_sizes, int n_in,
                              void* d_out, int out_size, void* d_ws, size_t ws_size,
                              hipStream_t stream) {
    const long long* edges = (const long long*)d_in[0];  // (2, E) int64
    const float*     att   = (const float*)d_in[1];      // (E,) f32
    const float*     X     = (const float*)d_in[3];      // (N, 128) f32
    float* out = (float*)d_out;

    long long E = (long long)in_sizes[1];
    int N = out_size / D_FEAT;

    size_t offB = (((size_t)N * 4) + 255) & ~(size_t)255;
    size_t need = offB + (size_t)E * 8;                  // off + csr_dst + csr_att

    if (ws_size >= need) {
        // CSR pull-based pipeline: ~6.4M u32 atomics instead of ~410M fp32 atomics
        unsigned* off     = (unsigned*)d_ws;
        int*      csr_dst = (int*)((char*)d_ws + offB);
        float*    csr_att = (float*)((char*)d_ws + offB + (size_t)E * 4);

        spmm_zero_u32<<<(N + 255) / 256, 256, 0, stream>>>(off, N);
        spmm_hist_kernel<<<4096, 256, 0, stream>>>(edges, off, E);
        spmm_scan_kernel<<<1, SCAN_T, 0, stream>>>(off, N);
        spmm_fill_kernel<<<4096, 256, 0, stream>>>(edges, att, off, csr_dst, csr_att, E);
        int gblocks = (N + WAVES_PER_BLOCK - 1) / WAVES_PER_BLOCK;  // one wave per node
        spmm_gather_kernel<<<gblocks, BLOCK, 0, stream>>>(off, csr_dst, csr_att, X, out, N);
    } else {
        // Fallback: atomic scatter
        int n4 = out_size / 4;
        spmm_zero_kernel<<<(n4 + 255) / 256, 256, 0, stream>>>((float4*)out, n4);
        spmm_scatter_kernel<<<2048, BLOCK, 0, stream>>>(edges, att, X, out, E);
    }
}